// CRF_70480413328058
// MI455X (gfx1250) — compile-verified
//
#include <hip/hip_runtime.h>

#define NUM_TAGS  32
#define START_TAG 30
#define STOP_TAG  31
#define NEG_INF   (-10000.0f)

#define LOG2E 1.4426950408889634f
#define LN2   0.6931471805599453f

typedef float v2f __attribute__((ext_vector_type(2)));
typedef float v4f __attribute__((ext_vector_type(4)));
typedef float v8f __attribute__((ext_vector_type(8)));

#define WAVES_PER_BLOCK 1
#define LDS_STRIDE 34   // floats per batch row in LDS; even => 8B-aligned ds_load_b64

// One wave handles a 16-batch tile for all S steps. The CRF recurrence is run
// entirely in the base-2 log domain (beta = alpha * log2e) so exp/log lower to
// bare v_exp_f32 / v_log_f32 with no scaling muls or denorm-guard sequences.
//
// beta state is kept in WMMA D layout: tile t in {0,1} covers tags 16t..16t+15;
//   VGPR r, lanes 0-15  -> (tag = r     + 16t, batch = lane)
//   VGPR r, lanes 16-31 -> (tag = r + 8 + 16t, batch = lane-16)
__global__ __launch_bounds__(WAVES_PER_BLOCK * 32)
void crf_forward_wmma(const float* __restrict__ feat,   // [B, S, T]
                      const float* __restrict__ trans,  // [T, T]  trans[i][j] = j -> i
                      float* __restrict__ out,          // [B]
                      int B, int S)
{
    __shared__ float lds[WAVES_PER_BLOCK * 16 * LDS_STRIDE];

    const int lane = threadIdx.x & 31;
    const int wave = threadIdx.x >> 5;
    const int half = lane >> 4;     // 0: lanes 0-15, 1: lanes 16-31
    const int col  = lane & 15;     // batch column within the 16-batch tile
    const int tileBase = (blockIdx.x * WAVES_PER_BLOCK + wave) * 16;
    const int b = tileBase + col;   // this lane's batch row (B divisible by 16*WAVES)

    float* wlds = lds + wave * 16 * LDS_STRIDE;

    // ---- Preload constant A = exp(trans) = exp2(trans*log2e), WMMA A-layout ----
    // tile t, K-chunk c (j = 4c..4c+3):
    //   VGPR0: lanes0-15 (i=lane, k=4c),   lanes16-31 (i=lane-16, k=4c+2)
    //   VGPR1: k = 4c+1 / 4c+3
    v2f Amat[2][8];
    #pragma unroll
    for (int t = 0; t < 2; ++t) {
        const int i = col + 16 * t;
        #pragma unroll
        for (int c = 0; c < 8; ++c) {
            const int j0 = 4 * c + 2 * half;           // even -> 8B aligned
            v2f e = *(const v2f*)(trans + i * NUM_TAGS + j0);
            Amat[t][c].x = __builtin_amdgcn_exp2f(e.x * LOG2E);
            Amat[t][c].y = __builtin_amdgcn_exp2f(e.y * LOG2E);
        }
    }

    // STOP-row transition scores (base-2 scaled) in D layout for the final LSE
    float tstop[2][8];
    #pragma unroll
    for (int t = 0; t < 2; ++t)
        #pragma unroll
        for (int r = 0; r < 8; ++r)
            tstop[t][r] = trans[STOP_TAG * NUM_TAGS + (r + 8 * half + 16 * t)] * LOG2E;

    // ---- beta0 = alpha0*log2e: NEG_INF scale everywhere, 0 at START_TAG ----
    // START_TAG = 30 -> tile1, r + 8*half = 14 -> r=6, half=1
    float beta[2][8];
    #pragma unroll
    for (int t = 0; t < 2; ++t)
        #pragma unroll
        for (int r = 0; r < 8; ++r)
            beta[t][r] = NEG_INF * LOG2E;
    if (half == 1) beta[1][6] = 0.0f;

    const float* featRow = feat + (size_t)b * S * NUM_TAGS;

    for (int s = 0; s < S; ++s) {
        // Emissions for this step, loaded straight into D layout.
        // Per lane: tags {8*half .. 8*half+7} (tile0) and +16 (tile1), 16B aligned.
        const float* fp = featRow + (size_t)s * NUM_TAGS + 8 * half;
        v4f em0a = *(const v4f*)(fp);        // tile0, r=0..3
        v4f em0b = *(const v4f*)(fp + 4);    // tile0, r=4..7
        v4f em1a = *(const v4f*)(fp + 16);   // tile1, r=0..3
        v4f em1b = *(const v4f*)(fp + 20);   // tile1, r=4..7
        if (s + 1 < S) __builtin_prefetch(fp + NUM_TAGS, 0, 0);

        // Running per-batch max over all 32 tags: per-lane max + one half-swap.
        float m = beta[0][0];
        #pragma unroll
        for (int t = 0; t < 2; ++t)
            #pragma unroll
            for (int r = 0; r < 8; ++r)
                m = fmaxf(m, beta[t][r]);
        m = fmaxf(m, __shfl_xor(m, 16, 32));

        // v = exp2(beta - m), scatter into LDS as [batch][tag] (D -> B transpose)
        #pragma unroll
        for (int t = 0; t < 2; ++t)
            #pragma unroll
            for (int r = 0; r < 8; ++r) {
                const int j = r + 8 * half + 16 * t;
                wlds[col * LDS_STRIDE + j] = __builtin_amdgcn_exp2f(beta[t][r] - m);
            }
        asm volatile("s_wait_dscnt 0" ::: "memory");   // stores visible before B loads

        // D = E x v : 8 chained K=4 fp32 WMMAs per 16-tag tile
        v8f acc0 = {0.f, 0.f, 0.f, 0.f, 0.f, 0.f, 0.f, 0.f};
        v8f acc1 = {0.f, 0.f, 0.f, 0.f, 0.f, 0.f, 0.f, 0.f};
        #pragma unroll
        for (int c = 0; c < 8; ++c) {
            // B chunk in WMMA B-layout via one b64 load:
            //   VGPR0 = v[4c + 2*half][col], VGPR1 = v[4c+1 + 2*half][col]
            v2f Bv = *(const v2f*)(wlds + col * LDS_STRIDE + 4 * c + 2 * half);
            acc0 = __builtin_amdgcn_wmma_f32_16x16x4_f32(
                       false, Amat[0][c], false, Bv, (short)0, acc0, false, false);
            acc1 = __builtin_amdgcn_wmma_f32_16x16x4_f32(
                       false, Amat[1][c], false, Bv, (short)0, acc1, false, false);
        }
        asm volatile("s_wait_dscnt 0" ::: "memory");   // B loads done before next overwrite

        // beta' = log2(D) + m + emit*log2e  (bare v_log_f32 + one fma per element)
        #pragma unroll
        for (int r = 0; r < 8; ++r) {
            const float e0 = (r < 4) ? em0a[r] : em0b[r - 4];
            const float e1 = (r < 4) ? em1a[r] : em1b[r - 4];
            beta[0][r] = __builtin_amdgcn_logf(acc0[r]) + fmaf(e0, LOG2E, m);
            beta[1][r] = __builtin_amdgcn_logf(acc1[r]) + fmaf(e1, LOG2E, m);
        }
    }

    // ---- log2 Z[b] = lse2_i(beta[i,b] + tstop2[i]);  logZ = ln2 * log2Z ----
    float vmax = beta[0][0] + tstop[0][0];
    #pragma unroll
    for (int t = 0; t < 2; ++t)
        #pragma unroll
        for (int r = 0; r < 8; ++r)
            vmax = fmaxf(vmax, beta[t][r] + tstop[t][r]);
    vmax = fmaxf(vmax, __shfl_xor(vmax, 16, 32));

    float ssum = 0.0f;
    #pragma unroll
    for (int t = 0; t < 2; ++t)
        #pragma unroll
        for (int r = 0; r < 8; ++r)
            ssum += __builtin_amdgcn_exp2f(beta[t][r] + tstop[t][r] - vmax);
    ssum += __shfl_xor(ssum, 16, 32);

    const float logz = (__builtin_amdgcn_logf(ssum) + vmax) * LN2;
    if (half == 0) out[b] = logz;
}

extern "C" void kernel_launch(void* const* d_in, const int* in_sizes, int n_in,
                              void* d_out, int out_size, void* d_ws, size_t ws_size,
                              hipStream_t stream) {
    const float* feat  = (const float*)d_in[0];   // [B, S, T] fp32
    const float* trans = (const float*)d_in[1];   // [T, T] fp32
    float* out = (float*)d_out;                   // [B] fp32

    const int B = out_size;                               // 2048
    const int S = in_sizes[0] / (B * NUM_TAGS);           // 512

    const int batchesPerBlock = 16 * WAVES_PER_BLOCK;
    dim3 grid(B / batchesPerBlock);                       // 128 blocks
    dim3 block(WAVES_PER_BLOCK * 32);                     // 1 wave each
    crf_forward_wmma<<<grid, block, 0, stream>>>(feat, trans, out, B, S);
}